// Embed_30520037606029
// MI455X (gfx1250) — compile-verified
//
#include <hip/hip_runtime.h>

typedef float v2f __attribute__((ext_vector_type(2)));
typedef float v8f __attribute__((ext_vector_type(8)));

#define B_DIM   16
#define L_DIM   128
#define LOC_DIM 1024
#define EMB_DIM 16

__global__ __launch_bounds__(256) void stlstm_embed_wmma(
    const int*   __restrict__ traj_loc,   // (B, L)
    const float* __restrict__ mat2,       // (LOC_MAX, LOC_MAX)
    const float* __restrict__ vec,        // (B, L)
    const int*   __restrict__ traj_len,   // (B,)
    const float* __restrict__ emb_su_w,   // (2, EMB)
    const float* __restrict__ emb_sl_w,   // (2, EMB)
    const float* __restrict__ emb_tu_w,   // (2, EMB)
    const float* __restrict__ emb_tl_w,   // (2, EMB)
    float*       __restrict__ out)        // (B, L, LOC_MAX, EMB)
{
    const float SU = 100.0f;
    const float TU = 24.0f;

    const int bl = blockIdx.x;        // 0 .. B*L-1
    const int b  = bl >> 7;           // / L_DIM
    const int l  = bl & (L_DIM - 1);

    const bool valid = (l < traj_len[b]);
    const int  mask  = valid ? 1 : 0;
    const int  row   = traj_loc[bl] - 1;
    const float t    = vec[bl];

    const int  lane = threadIdx.x & 31;
    const int  wave = threadIdx.x >> 5;   // 0..7
    const int  e    = lane & 15;
    const bool lo   = lane < 16;

    // Rank-2 factorization: out[k,e] = ds[k]*u[e] + 1*v[e]
    const float esl = emb_sl_w[mask * EMB_DIM + e];
    const float esu = emb_su_w[mask * EMB_DIM + e];
    const float etl = emb_tl_w[mask * EMB_DIM + e];
    const float etu = emb_tu_w[mask * EMB_DIM + e];

    const float u        = (esu - esl) * (1.0f / SU);
    const float timeTerm = (etl * (TU - t) + etu * t) * (1.0f / TU);
    const float v        = esl + timeTerm;

    // B matrix (4x16 f32): lanes 0-15 = rows K=0 (u), K=1 (v); lanes 16-31
    // = rows K=2,3 (zero padding).
    v2f bm;
    bm.x = lo ? u : 0.0f;
    bm.y = lo ? v : 0.0f;

    // A-matrix constants: A[M,K=1] = 1 (lanes 0-15); multiplicative mask
    // zeroes ds for invalid rows and for the K=2,3 padding lanes without
    // touching EXEC (keeps loads clauseable, no saveexec per tile).
    const float amask = (valid && lo) ? 1.0f : 0.0f;
    const float ay    = lo ? 1.0f : 0.0f;

    // Per-lane base pointers; all further addressing is immediate offsets.
    const float* rowp = mat2 + (size_t)row * LOC_DIM + wave * 128 + e;
    float* basep = out + (size_t)bl * (LOC_DIM * EMB_DIM)
                       + (size_t)(wave * 128 + (lo ? 0 : 8)) * EMB_DIM + e;

    // Stage all 8 row fragments first -> single 8-wide load clause.
    float ds[8];
    #pragma unroll
    for (int ti = 0; ti < 8; ++ti)
        ds[ti] = rowp[ti * 16];           // imm offset ti*64 bytes

    #pragma unroll
    for (int ti = 0; ti < 8; ++ti) {
        v2f a;
        a.x = ds[ti] * amask;             // A[M,K=0] = masked ds
        a.y = ay;                         // A[M,K=1] = 1
        v8f c = {0.f, 0.f, 0.f, 0.f, 0.f, 0.f, 0.f, 0.f};
        c = __builtin_amdgcn_wmma_f32_16x16x4_f32(
                /*neg_a=*/false, a, /*neg_b=*/false, bm,
                /*c_mod=*/0, c, /*reuse_a=*/false, /*reuse_b=*/false);

        // D layout: VGPR r -> M = r / r+8, N = lane%16. Element (k,e) at
        // basep + (ti*16 + r)*EMB floats -> imm offset ti*1024 + r*64 bytes.
        #pragma unroll
        for (int r = 0; r < 8; ++r)
            __builtin_nontemporal_store(c[r], basep + (ti * 16 + r) * EMB_DIM);
    }
}

extern "C" void kernel_launch(void* const* d_in, const int* in_sizes, int n_in,
                              void* d_out, int out_size, void* d_ws, size_t ws_size,
                              hipStream_t stream) {
    const int*   traj_loc = (const int*)  d_in[0];
    const float* mat2     = (const float*)d_in[1];
    const float* vec      = (const float*)d_in[2];
    const int*   traj_len = (const int*)  d_in[3];
    const float* emb_su_w = (const float*)d_in[4];
    const float* emb_sl_w = (const float*)d_in[5];
    const float* emb_tu_w = (const float*)d_in[6];
    const float* emb_tl_w = (const float*)d_in[7];
    float* out = (float*)d_out;

    dim3 grid(B_DIM * L_DIM);   // 2048 workgroups, one (b,l) 1024x16 tile each
    dim3 block(256);            // 8 wave32s
    hipLaunchKernelGGL(stlstm_embed_wmma, grid, block, 0, stream,
                       traj_loc, mat2, vec, traj_len,
                       emb_su_w, emb_sl_w, emb_tu_w, emb_tl_w, out);
}